// Predictor_33440615367337
// MI455X (gfx1250) — compile-verified
//
#include <hip/hip_runtime.h>
#include <math.h>

#define B_  16
#define U_  2048
#define D_  512
#define KT_ 1536   // 3 * 512

typedef float v2f __attribute__((ext_vector_type(2)));
typedef float v8f __attribute__((ext_vector_type(8)));

// ---------------------------------------------------------------------------
// gfx1250 async global -> LDS (no data VGPRs, tracked by ASYNCcnt)
// ---------------------------------------------------------------------------
__device__ __forceinline__ void async_b128(uint32_t lds_addr, const void* gaddr) {
  asm volatile("global_load_async_to_lds_b128 %0, %1, off"
               :: "v"(lds_addr), "v"(gaddr)
               : "memory");
}
__device__ __forceinline__ void wait_async0() {
#if __has_builtin(__builtin_amdgcn_s_wait_asynccnt)
  __builtin_amdgcn_s_wait_asynccnt(0);
#else
  asm volatile("s_wait_asynccnt 0x0" ::: "memory");
#endif
}
__device__ __forceinline__ uint32_t lds_addr_of(const void* p) {
  // generic shared pointer: low 32 bits are the LDS byte offset
  return (uint32_t)(uintptr_t)p;
}

// ---------------------------------------------------------------------------
// Kernel 0: repack conv_w (D_out, D_in, 3) -> Wre[chunk][d][kc], kc-minor,
// where global K = chunk*64 + kc = k*512 + c.  K-minor makes the WMMA A
// fragment a contiguous float2 in LDS (ds_load_b64, no register repacking).
// ---------------------------------------------------------------------------
__global__ void repack_w(const float* __restrict__ conv_w, float* __restrict__ Wre) {
    int idx = blockIdx.x * blockDim.x + threadIdx.x;   // 0 .. 786431
    if (idx >= KT_ * D_) return;
    int kc    = idx & 63;
    int d     = (idx >> 6) & (D_ - 1);
    int chunk = idx >> 15;                 // 0..23
    int K = chunk * 64 + kc;
    int c = K & (D_ - 1);
    int k = K >> 9;
    Wre[idx] = conv_w[d * KT_ + c * 3 + k];
}

// ---------------------------------------------------------------------------
// Async staging of one K-chunk: A tile 128d x 64kc, B tile 128u x 64c.
// Every transfer is a contiguous 16B quantum straight into LDS.
// ---------------------------------------------------------------------------
__device__ __forceinline__ void stage_chunk(
    const float* __restrict__ Wre, const float* __restrict__ xsb,
    int d0, int chunk, int u0, int tid,
    uint32_t aLds, uint32_t bLds, float* __restrict__ bPtr)
{
  const int k  = chunk >> 3;
  const int c0 = (chunk & 7) << 6;
  const float* aSrc = Wre + ((size_t)chunk * D_ + d0) * 64;
  // A: 128 d-rows x 16 float4 = 2048 transfers, 8 per thread. LDS row stride 68 f.
  #pragma unroll
  for (int t = 0; t < 8; ++t) {
    int q   = tid + t * 256;
    int dr  = q >> 4;          // 0..127
    int kc4 = q & 15;          // float4 index within the 64-wide K chunk
    async_b128(aLds + (uint32_t)(dr * 272 + kc4 * 16),
               aSrc + (size_t)dr * 64 + kc4 * 4);
  }
  // B: 128 u-rows x 16 float4 = 2048 transfers, 8 per thread. LDS row stride 68 f.
  #pragma unroll
  for (int t = 0; t < 8; ++t) {
    int q   = tid + t * 256;
    int u   = q >> 4;
    int c4  = q & 15;
    int row = u0 + u + k - 1;
    if (row >= 0 && row < U_) {
      async_b128(bLds + (uint32_t)(u * 272 + c4 * 16),
                 xsb + (size_t)row * D_ + c0 + c4 * 4);
    } else {
      *(float4*)&bPtr[u * 68 + c4 * 4] = make_float4(0.f, 0.f, 0.f, 0.f);
    }
  }
}

// ---------------------------------------------------------------------------
// Kernel 1: fused conv-GEMM + bias + relu + lin_w dot + sigmoid + mask -> alpha.
// FP32 WMMA 16x16x4. Grid: (U/128, B), 256 threads = 8 waves = 2(d) x 4(u).
// Wave owns 4 d-subtiles x 2 u-subtiles. Async double-buffered LDS pipeline.
// ---------------------------------------------------------------------------
__global__ __launch_bounds__(256) void conv_alpha(
    const float* __restrict__ xs, const float* __restrict__ Wre,
    const float* __restrict__ conv_b, const float* __restrict__ lin_w,
    const float* __restrict__ lin_b, const int* __restrict__ xlens,
    float* __restrict__ alpha)
{
  __shared__ float A_lds[2][128 * 68];   // [buf][d_local][kc], pad 64->68
  __shared__ float B_lds[2][128 * 68];   // [buf][u_local][c],  pad 64->68
  __shared__ float red[256];

  const int tid  = threadIdx.x;
  const int b    = blockIdx.y;
  const int u0   = blockIdx.x * 128;
  const int w    = tid >> 5, lane = tid & 31;
  const int wd   = w >> 2, wu = w & 3;
  const int half = lane >> 4, m = lane & 15;

  const float* xsb = xs + (size_t)b * U_ * D_;
  const uint32_t aL[2] = { lds_addr_of(&A_lds[0][0]), lds_addr_of(&A_lds[1][0]) };
  const uint32_t bL[2] = { lds_addr_of(&B_lds[0][0]), lds_addr_of(&B_lds[1][0]) };

  float psum0 = 0.f, psum1 = 0.f;

  for (int dblk = 0; dblk < 4; ++dblk) {
    const int d0 = dblk * 128;
    v8f acc[4][2] = {};

    stage_chunk(Wre, xsb, d0, 0, u0, tid, aL[0], bL[0], &B_lds[0][0]);
    wait_async0();
    __syncthreads();

    for (int chunk = 0; chunk < 24; ++chunk) {
      const int cur = chunk & 1;
      if (chunk + 1 < 24)
        stage_chunk(Wre, xsb, d0, chunk + 1, u0, tid,
                    aL[cur ^ 1], bL[cur ^ 1], &B_lds[cur ^ 1][0]);

      const float* Ab = &A_lds[cur][0];
      const float* Bb = &B_lds[cur][0];
      #pragma unroll
      for (int step = 0; step < 16; ++step) {
        const int kk = step * 4;
        v2f bf[2];
        #pragma unroll
        for (int g = 0; g < 2; ++g) {
          const int ur = (wu * 2 + g) * 16 + m;
          bf[g].x = Bb[ur * 68 + kk + half];
          bf[g].y = Bb[ur * 68 + kk + 2 + half];
        }
        #pragma unroll
        for (int s = 0; s < 4; ++s) {
          const int dl = wd * 64 + s * 16 + m;
          // contiguous float2: K = kk + 2*half + {0,1}
          v2f af = *(const v2f*)&Ab[dl * 68 + kk + 2 * half];
          #pragma unroll
          for (int g = 0; g < 2; ++g)
            acc[s][g] = __builtin_amdgcn_wmma_f32_16x16x4_f32(
                false, af, false, bf[g], (short)0, acc[s][g], false, false);
        }
      }
      wait_async0();
      __syncthreads();
    }

    // epilogue: bias + relu + dot with lin_w
    #pragma unroll
    for (int s = 0; s < 4; ++s) {
      const int dbase = d0 + wd * 64 + s * 16 + 8 * half;
      #pragma unroll
      for (int r = 0; r < 8; ++r) {
        const int d = dbase + r;
        const float cb = conv_b[d], lw = lin_w[d];
        float h0 = acc[s][0][r] + cb;
        float h1 = acc[s][1][r] + cb;
        psum0 += fmaxf(h0, 0.f) * lw;
        psum1 += fmaxf(h1, 0.f) * lw;
      }
    }
  }

  // lanes L and L+16 share the same u column
  psum0 += __shfl_xor(psum0, 16, 32);
  psum1 += __shfl_xor(psum1, 16, 32);
  __syncthreads();
  if (half == 0) {
    red[w * 32 + m]      = psum0;
    red[w * 32 + 16 + m] = psum1;
  }
  __syncthreads();
  if (tid < 128) {
    const int wuu = tid >> 5;
    const int g   = (tid >> 4) & 1;
    const int un  = tid & 15;
    float s = red[wuu * 32 + g * 16 + un] + red[(4 + wuu) * 32 + g * 16 + un];
    float z = s + lin_b[0];
    float a = 1.f / (1.f + __expf(-z));
    const int ug = u0 + (wuu * 2 + g) * 16 + un;
    if (ug >= xlens[b]) a = 0.f;
    alpha[b * U_ + ug] = a;
  }
}

// ---------------------------------------------------------------------------
// Kernel 2: per-batch alpha prefix scan -> firing schedule (L, R, endt, nf).
// ---------------------------------------------------------------------------
__global__ __launch_bounds__(256) void cif_scan(
    const float* __restrict__ alpha, const int* __restrict__ ylens,
    float* __restrict__ Lc, float* __restrict__ Rc,
    int* __restrict__ endt, int* __restrict__ nf,
    float* __restrict__ sum_alpha_out)
{
  __shared__ float sA[U_];
  __shared__ float st[256];
  const int b = blockIdx.x;
  const int t = threadIdx.x;

  for (int i = 0; i < 8; ++i) sA[t + i * 256] = alpha[b * U_ + t + i * 256];
  __syncthreads();

  float v[8]; float run = 0.f;
  #pragma unroll
  for (int i = 0; i < 8; ++i) { run += sA[t * 8 + i]; v[i] = run; }
  st[t] = run;
  __syncthreads();
  for (int off = 1; off < 256; off <<= 1) {
    float x = (t >= off) ? st[t - off] : 0.f;
    __syncthreads();
    st[t] += x;
    __syncthreads();
  }
  const float total = st[255];
  const float base  = (t == 0) ? 0.f : st[t - 1];
  #pragma unroll
  for (int i = 0; i < 8; ++i) sA[t * 8 + i] = v[i] + base;
  __syncthreads();
  if (t == 0) sum_alpha_out[b] = total;

  float beta = total / (float)ylens[b] - 1e-4f;
  if (!(beta > 1e-6f)) beta = 1e-6f;
  const float invb = 1.f / beta;

  #pragma unroll
  for (int i = 0; i < 8; ++i) {
    int ti   = t * 8 + i;
    float Ap = (ti == 0) ? 0.f : sA[ti - 1];
    float At = sA[ti];
    int kp = (int)floorf(Ap * invb);
    int kt = (int)floorf(At * invb);
    Lc[b * U_ + ti] = beta - (Ap - beta * (float)kp);
    if (kt > kp) {
      int j = kp;
      if (j < 512) {
        endt[b * 512 + j] = ti;
        Rc[b * 512 + j]   = At - beta * (float)(kp + 1);
      }
    }
    if (ti == U_ - 1) nf[b] = (kt > 512) ? 512 : kt;
  }
}

// ---------------------------------------------------------------------------
// Kernel 3: segmented weighted sums -> h_cif rows (compaction implicit).
// ---------------------------------------------------------------------------
__global__ __launch_bounds__(256) void cif_emit(
    const float* __restrict__ xs, const float* __restrict__ Lc,
    const float* __restrict__ Rc, const int* __restrict__ endt,
    const int* __restrict__ nf, float* __restrict__ out, int u_max)
{
  const int b = blockIdx.y, j = blockIdx.x;
  if (j >= nf[b]) return;
  const int end    = endt[b * 512 + j];
  const int start  = (j == 0) ? 0 : endt[b * 512 + j - 1];
  const float firstc = (j == 0) ? Lc[b * U_ + start] : Rc[b * 512 + j - 1];
  const int d = threadIdx.x * 2;
  float ax = 0.f, ay = 0.f;
  for (int tt = start; tt <= end; ++tt) {
    float c = (tt == start) ? firstc : Lc[b * U_ + tt];
    const float2 x = *(const float2*)&xs[((size_t)b * U_ + tt) * D_ + d];
    ax = fmaf(c, x.x, ax);
    ay = fmaf(c, x.y, ay);
  }
  *(float2*)&out[((size_t)b * u_max + j) * D_ + d] = make_float2(ax, ay);
}

// ---------------------------------------------------------------------------
extern "C" void kernel_launch(void* const* d_in, const int* in_sizes, int n_in,
                              void* d_out, int out_size, void* d_ws, size_t ws_size,
                              hipStream_t stream) {
  (void)in_sizes; (void)n_in; (void)ws_size;
  const float* xs     = (const float*)d_in[0];
  const int*   xlens  = (const int*)d_in[1];
  const int*   ylens  = (const int*)d_in[2];
  const float* conv_w = (const float*)d_in[3];
  const float* conv_b = (const float*)d_in[4];
  const float* lin_w  = (const float*)d_in[5];
  const float* lin_b  = (const float*)d_in[6];
  float* out = (float*)d_out;

  const int u_max = (out_size - B_) / (B_ * D_);

  float* Wre   = (float*)d_ws;                 // 786432 f
  float* alpha = Wre + (size_t)KT_ * D_;       // 32768 f
  float* Lc    = alpha + B_ * U_;              // 32768 f
  float* Rc    = Lc + B_ * U_;                 // 8192 f
  int*   endt  = (int*)(Rc + B_ * 512);        // 8192 i
  int*   nf    = endt + B_ * 512;              // 16 i

  hipMemsetAsync(d_out, 0, (size_t)out_size * sizeof(float), stream);

  repack_w<<<(KT_ * D_ + 255) / 256, 256, 0, stream>>>(conv_w, Wre);
  conv_alpha<<<dim3(U_ / 128, B_), 256, 0, stream>>>(xs, Wre, conv_b, lin_w,
                                                     lin_b, xlens, alpha);
  float* sum_out = out + (size_t)B_ * u_max * D_;
  cif_scan<<<B_, 256, 0, stream>>>(alpha, ylens, Lc, Rc, endt, nf, sum_out);
  cif_emit<<<dim3(u_max, B_), 256, 0, stream>>>(xs, Lc, Rc, endt, nf, out, u_max);
}